// CTCLayer_807453852026
// MI455X (gfx1250) — compile-verified
//
#include <hip/hip_runtime.h>

// Problem dims (fixed by the reference): B=512, T=512, C=128, L=64.
#define CTC_B 512
#define CTC_T 512
#define CTC_C 128
#define CTC_L 64
#define CTC_S (2 * CTC_L + 1)   // 129 extended-label slots
#define CTC_BLANK (CTC_C - 1)   // 127
#define NEG_INF (-1e30f)
#define CTC_EPS 1e-7f

#define NBUF 8   // row ring buffers (power of 2)
#define PFD  6   // async prefetch depth (steady-state wait = PFD-2)

#define LOG2E 1.44269504088896340736f  // ln -> log2 scale
#define LN2   0.69314718055994530942f  // log2 -> ln scale

// Single-instruction hardware transcendentals (TRANS pipe):
//   v_exp_f32 computes 2^x, v_log_f32 computes log2(x).
__device__ __forceinline__ float hw_exp2(float x) { return __builtin_amdgcn_exp2f(x); }
__device__ __forceinline__ float hw_log2(float x) { return __builtin_amdgcn_logf(x); }

// Generic LDS pointer -> 32-bit LDS byte address (low 32 bits of the
// LDS-aperture flat address are the LDS offset on gfx1250).
__device__ __forceinline__ unsigned lds_addr_u32(const void* p) {
  return (unsigned)(unsigned long long)p;
}

// Per-lane async DMA of 16 bytes global -> LDS (ASYNCcnt-tracked).
__device__ __forceinline__ void async_b128_to_lds(unsigned lds_dst,
                                                  unsigned long long gsrc) {
  asm volatile("global_load_async_to_lds_b128 %0, %1, off"
               :: "v"(lds_dst), "v"(gsrc)
               : "memory");
}

// ASYNCcnt completes in order: waiting for <= N proves all but the N newest
// outstanding async loads have fully landed in LDS.
__device__ __forceinline__ void wait_async4() { asm volatile("s_wait_asynccnt 4" ::: "memory"); }
__device__ __forceinline__ void wait_async3() { asm volatile("s_wait_asynccnt 3" ::: "memory"); }
__device__ __forceinline__ void wait_async2() { asm volatile("s_wait_asynccnt 2" ::: "memory"); }
__device__ __forceinline__ void wait_async1() { asm volatile("s_wait_asynccnt 1" ::: "memory"); }
__device__ __forceinline__ void wait_async0() { asm volatile("s_wait_asynccnt 0" ::: "memory"); }

// Fused 3-way logaddexp using v_max3_f32 + 3x v_exp_f32 + v_log_f32.
// Safe with the -1e30 "-inf" surrogate: exp2(-1.44e30) flushes to 0.
__device__ __forceinline__ float lse3(float a0, float a1, float a2) {
  float m = fmaxf(fmaxf(a0, a1), a2);                 // v_max3_f32
  float s = hw_exp2((a0 - m) * LOG2E)
          + hw_exp2((a1 - m) * LOG2E)
          + hw_exp2((a2 - m) * LOG2E);
  return fmaf(LN2, hw_log2(s), m);
}

// One workgroup per batch element. 160 threads = 5 wave32:
//   all waves   : full-row sum redundantly per wave (no reduction barrier)
//   threads<129 : DP slot s = tid
//   wave 4      : also issues the async row prefetches
__global__ __launch_bounds__(160) void ctc_loss_kernel(
    const long long* __restrict__ y_true,   // [B, L] int64 labels in [0,127)
    const float* __restrict__ y_pred,       // [B, T, C] fp32 probabilities
    float* __restrict__ out) {              // [B] fp32 NLL
  __shared__ __align__(16) float rowbuf[NBUF][CTC_C];  // 8 x 512B async ring
  __shared__ float alpha[2][CTC_S + 3];                // ping-pong alpha

  const int b    = blockIdx.x;
  const int tid  = threadIdx.x;
  const int lane = tid & 31;
  const int wave = tid >> 5;

  const float* rowbase = y_pred + (size_t)b * CTC_T * CTC_C;

  // Extended label: blank at even s, label at odd s.
  // skip (s-2 -> s) allowed iff ext[s] != blank && ext[s] != ext[s-2].
  int  ext_s = CTC_BLANK;
  bool skip  = false;
  const bool isS = tid < CTC_S;
  if (isS && (tid & 1)) {
    ext_s = (int)y_true[(size_t)b * CTC_L + (tid >> 1)];
    if (tid >= 2) {
      int ext_m2 = (int)y_true[(size_t)b * CTC_L + (tid >> 1) - 1];
      skip = (ext_s != ext_m2);
    }
  }

  // ---- prologue: fill the async pipeline with rows 0..PFD-1 ----
  if (wave == 4) {
    #pragma unroll
    for (int r = 0; r < PFD; ++r) {
      async_b128_to_lds(lds_addr_u32(&rowbuf[r][0]) + (unsigned)(lane * 16),
                        (unsigned long long)(rowbase + (size_t)r * CTC_C)
                          + (unsigned long long)lane * 16);
    }
  }
  wait_async4();     // rows 0 and 1 have landed (in-order completion)
  __syncthreads();

  // Emission for t=0, computed from row 0 (off the DP critical chain).
  // e[s] = ln(p[ext_s]+eps) - ln(sum_c p + C*eps) = LN2*(log2(..) - log2(..))
  float e_cur = 0.0f;
  {
    const float4 r4 = *reinterpret_cast<const float4*>(&rowbuf[0][lane * 4]);
    float v = (r4.x + r4.y) + (r4.z + r4.w);
    #pragma unroll
    for (int off = 16; off > 0; off >>= 1)
      v += __shfl_xor(v, off, 32);
    float l2sum = hw_log2(v + (float)CTC_C * CTC_EPS);
    if (isS) e_cur = LN2 * (hw_log2(rowbuf[0][ext_s] + CTC_EPS) - l2sum);
  }

  for (int t = 0; t < CTC_T; ++t) {
    // Keep the async pipeline PFD rows deep.
    if (wave == 4 && (t + PFD) < CTC_T) {
      const float* nsrc = rowbase + (size_t)(t + PFD) * CTC_C;
      async_b128_to_lds(
          lds_addr_u32(&rowbuf[(t + PFD) & (NBUF - 1)][0]) + (unsigned)(lane * 16),
          (unsigned long long)nsrc + (unsigned long long)lane * 16);
    }

    // Pipelined emission for step t+1 (independent of alpha): row t+1 is
    // guaranteed resident by last step's s_wait_asynccnt.
    float e_next = 0.0f;
    if (t + 1 < CTC_T) {
      const int nb = (t + 1) & (NBUF - 1);
      const float4 r4 = *reinterpret_cast<const float4*>(&rowbuf[nb][lane * 4]);
      float v = (r4.x + r4.y) + (r4.z + r4.w);
      #pragma unroll
      for (int off = 16; off > 0; off >>= 1)
        v += __shfl_xor(v, off, 32);
      float l2sum = hw_log2(v + (float)CTC_C * CTC_EPS);
      if (isS) e_next = LN2 * (hw_log2(rowbuf[nb][ext_s] + CTC_EPS) - l2sum);
    }

    // DP update (the serial critical chain): fused 3-way log-sum-exp.
    if (isS) {
      const int w = t & 1, r = w ^ 1;
      float nv;
      if (t == 0) {
        nv = (tid < 2) ? e_cur : NEG_INF;  // alpha0: s=0 (blank), s=1 (label0)
      } else {
        float a0 = alpha[r][tid];
        float a1 = (tid >= 1) ? alpha[r][tid - 1] : NEG_INF;
        float a2 = (tid >= 2 && skip) ? alpha[r][tid - 2] : NEG_INF;
        nv = lse3(a0, a1, a2) + e_cur;
      }
      alpha[w][tid] = nv;
    }

    // Prove rows t+1 and t+2 have landed (needed for next step's pipelined
    // emission), then one barrier covers alpha ping-pong WAR, ring WAR and
    // DMA-data visibility.
    if      (t + PFD < CTC_T) wait_async4();   // steady state: <= PFD-2
    else if (t + 5   < CTC_T) wait_async3();
    else if (t + 4   < CTC_T) wait_async2();
    else if (t + 3   < CTC_T) wait_async1();
    else                      wait_async0();
    __syncthreads();

    e_cur = e_next;
  }

  if (tid == 0) {
    const int w = (CTC_T - 1) & 1;
    float ll = lse3(alpha[w][CTC_S - 1], alpha[w][CTC_S - 2], NEG_INF);
    out[b] = -ll;
  }
}

extern "C" void kernel_launch(void* const* d_in, const int* in_sizes, int n_in,
                              void* d_out, int out_size, void* d_ws, size_t ws_size,
                              hipStream_t stream) {
  (void)in_sizes; (void)n_in; (void)out_size; (void)d_ws; (void)ws_size;
  const long long* y_true = (const long long*)d_in[0];  // int64 per reference
  const float*     y_pred = (const float*)d_in[1];      // fp32 [B,T,C]
  float*           out    = (float*)d_out;              // fp32 [B,1]
  ctc_loss_kernel<<<dim3(CTC_B), dim3(160), 0, stream>>>(y_true, y_pred, out);
}